// RNNLM_61469571940718
// MI455X (gfx1250) — compile-verified
//
#include <hip/hip_runtime.h>
#include <hip/hip_bf16.h>

// RNN LM: SEQ=128, BATCH=64, VOCAB=32000, EMB=15, HID=16, K = EMB+HID = 31 (pad 32)

typedef __attribute__((ext_vector_type(2))) float v2f;
typedef __attribute__((ext_vector_type(4))) float v4f;
typedef __attribute__((ext_vector_type(8))) float v8f;

#define SEQ   128
#define BATCH 64
#define VOCAB 32000
#define EMB   15
#define HID   16
#define KPAD  32        // 31 real K + 1 zero pad

// ---------------------------------------------------------------------------
// Kernel 1: sequential recurrence on a single workgroup.
// Computes h_t for all steps, materializes combined[t] = [we[ids_t], h_t, 0pad]
// as [128][64][32] fp32 into workspace (1 MB -> stays in L2 for kernel 2).
// ---------------------------------------------------------------------------
__global__ __launch_bounds__(256)
void rnn_recurrence(const int* __restrict__ ids,    // [128][64]
                    const float* __restrict__ we,   // [32000][15]
                    const float* __restrict__ i2h,  // [31][16]
                    const float* __restrict__ h0,   // [64][16]
                    float* __restrict__ comb)       // [128][64][32]
{
    __shared__ float sh[BATCH * HID];      // current hidden
    __shared__ float sc[BATCH * KPAD];     // combined for this step
    __shared__ float sw[31 * HID];         // i2h weights

    const int tid = threadIdx.x;           // 256 threads = 8 waves

    for (int i = tid; i < BATCH * HID; i += 256) sh[i] = h0[i];
    for (int i = tid; i < 31 * HID;    i += 256) sw[i] = i2h[i];
    __syncthreads();

    for (int t = 0; t < SEQ; ++t) {
        // build combined[t] in LDS
        for (int i = tid; i < BATCH * KPAD; i += 256) {
            const int b = i >> 5, col = i & 31;
            float v;
            if (col < EMB)            v = we[ids[t * BATCH + b] * EMB + col];
            else if (col < EMB + HID) v = sh[b * HID + (col - EMB)];
            else                      v = 0.0f;   // K=31 zero pad
            sc[i] = v;
        }
        __syncthreads();

        // spill combined[t] to global workspace (hot for kernel 2 -> keep RT)
        float* ct = comb + t * (BATCH * KPAD);
        for (int i = tid; i < BATCH * KPAD; i += 256) ct[i] = sc[i];

        // h_new = tanh(combined @ i2h): 1024 outputs, 4 per thread
        float hn[4];
        #pragma unroll
        for (int r = 0; r < 4; ++r) {
            const int o = tid + r * 256;       // 0..1023 = b*16 + j
            const int b = o >> 4, j = o & 15;
            float acc = 0.0f;
            #pragma unroll
            for (int k = 0; k < 31; ++k)
                acc = fmaf(sc[b * KPAD + k], sw[k * HID + j], acc);
            hn[r] = tanhf(acc);
        }
        __syncthreads();
        #pragma unroll
        for (int r = 0; r < 4; ++r) sh[tid + r * 256] = hn[r];
        __syncthreads();
    }
}

// ---------------------------------------------------------------------------
// Kernel 2: big GEMM  out[t] = exp(tanh(combined[t] @ i2o))  via
// V_WMMA_F32_16X16X4_F32.  Block = 256 threads = 8 waves.  Each wave owns one
// N-tile of 16 and all four M-tiles (M=64): 4 accumulators, 8 K-steps of 4.
// Block covers N=128 -> grid (250, 128).  Output stores are NON-TEMPORAL so
// the 1.05 GB stream does not evict i2o (4 MB, reread by every block) or comb
// (1 MB) from the 192 MB L2.  Emits deterministic per-block partial sums.
// ---------------------------------------------------------------------------
__global__ __launch_bounds__(256)
void rnn_bigmm(const float* __restrict__ comb,     // [128][64][32]
               const float* __restrict__ i2o,      // [31][32000]
               float* __restrict__ out,            // [128][64][32000]
               float* __restrict__ partials)       // [128][256] (250 used)
{
    const int t    = blockIdx.y;
    const int nblk = blockIdx.x;          // 0..249
    const int tid  = threadIdx.x;
    const int wave = tid >> 5;
    const int lane = tid & 31;

    __shared__ float sA[BATCH * KPAD];    // 8 KB: combined[t], shared by waves
    __shared__ float wsum[8];

    const float* ct = comb + t * (BATCH * KPAD);
    for (int i = tid; i < BATCH * KPAD; i += 256) sA[i] = ct[i];
    __syncthreads();

    const int nbase = nblk * 128 + wave * 16;   // this wave's N tile base
    const int ncol  = nbase + (lane & 15);      // global column for B/D frags
    const int ksel  = lane >> 4;                // 0: K pair {0,1}; 1: {2,3}
    const int mrow  = lane & 15;                // A-frag M within tile

    v8f acc[4] = {};                            // 4 M-tiles of C/D (16x16 f32)

    #pragma unroll
    for (int kk = 0; kk < 8; ++kk) {
        const int kb = kk * 4 + ksel * 2;       // this lane's K base

        // B fragment 4x16: VGPR0=B[kb][n], VGPR1=B[kb+1][n]; pad k==31 -> 0
        v2f bf;
        bf.x = i2o[kb * VOCAB + ncol];                            // kb <= 30
        bf.y = (kb + 1 < 31) ? i2o[(kb + 1) * VOCAB + ncol] : 0.0f;

        #pragma unroll
        for (int mt = 0; mt < 4; ++mt) {
            // A fragment 16x4: lane m = lane&15 (+tile), VGPR0=A[m][kb], VGPR1=A[m][kb+1]
            const int m = mt * 16 + mrow;
            v2f af;
            af.x = sA[m * KPAD + kb];
            af.y = sA[m * KPAD + kb + 1];
            acc[mt] = __builtin_amdgcn_wmma_f32_16x16x4_f32(
                false, af, false, bf, (short)0, acc[mt], false, false);
        }
    }

    // epilogue: v = exp(tanh(c)); write unnormalized (NT); accumulate lane sum
    float lsum = 0.0f;
    const int  mhi   = (lane >> 4) * 8;         // lanes 16..31 hold rows M+8
    const long obase = (long)t * BATCH * VOCAB;
    #pragma unroll
    for (int mt = 0; mt < 4; ++mt) {
        #pragma unroll
        for (int r = 0; r < 8; ++r) {
            float v = __expf(tanhf(acc[mt][r]));
            lsum += v;
            const int M = mt * 16 + r + mhi;
            __builtin_nontemporal_store(v, &out[obase + (long)M * VOCAB + ncol]);
        }
    }

    // deterministic wave reduction (fixed butterfly order)
    #pragma unroll
    for (int off = 16; off > 0; off >>= 1)
        lsum += __shfl_xor(lsum, off, 32);
    if (lane == 0) wsum[wave] = lsum;
    __syncthreads();
    if (tid == 0) {
        float s = 0.0f;
        #pragma unroll
        for (int w = 0; w < 8; ++w) s += wsum[w];
        partials[t * 256 + nblk] = s;           // fixed slot per block
    }
}

// ---------------------------------------------------------------------------
// Kernel 3: per-step deterministic reduction of 250 partials -> 1/sum[t]
// ---------------------------------------------------------------------------
__global__ void rnn_reduce(const float* __restrict__ partials,
                           float* __restrict__ invs)
{
    const int t = threadIdx.x;                  // blockDim = 128
    float s = 0.0f;
    for (int i = 0; i < 250; ++i) s += partials[t * 256 + i];
    invs[t] = 1.0f / s;
}

// ---------------------------------------------------------------------------
// Kernel 4: scale pass  out[t] *= 1/sum[t].  Pure stream: NT load (last use)
// + NT store, float4 wide.
// ---------------------------------------------------------------------------
__global__ __launch_bounds__(256)
void rnn_scale(float* __restrict__ out, const float* __restrict__ invs)
{
    const int t = blockIdx.y;
    const float s = invs[t];
    const long base = (long)t * BATCH * VOCAB;      // 2,048,000 floats per t
    const int  i    = (blockIdx.x * 256 + threadIdx.x) * 4;
    v4f* p = reinterpret_cast<v4f*>(out + base + i);
    v4f v = __builtin_nontemporal_load(p);
    v *= s;
    __builtin_nontemporal_store(v, p);
}

// ---------------------------------------------------------------------------
extern "C" void kernel_launch(void* const* d_in, const int* in_sizes, int n_in,
                              void* d_out, int out_size, void* d_ws, size_t ws_size,
                              hipStream_t stream)
{
    (void)in_sizes; (void)n_in; (void)out_size; (void)ws_size;

    const int*   ids = (const int*)  d_in[0];   // [128][64]
    const float* we  = (const float*)d_in[1];   // [32000][15]
    const float* i2h = (const float*)d_in[2];   // [31][16]
    const float* i2o = (const float*)d_in[3];   // [31][32000]
    const float* h0  = (const float*)d_in[4];   // [64][16]
    float* out = (float*)d_out;                 // [128][64][32000]

    // workspace layout (floats): comb | partials | invs
    float* ws       = (float*)d_ws;
    float* comb     = ws;                        // 128*64*32  = 262144 floats (1 MB)
    float* partials = comb + SEQ * BATCH * KPAD; // 128*256    = 32768 floats
    float* invs     = partials + SEQ * 256;      // 128 floats

    rnn_recurrence<<<1, 256, 0, stream>>>(ids, we, i2h, h0, comb);

    dim3 g2(VOCAB / 128, SEQ);                  // (250, 128)
    rnn_bigmm<<<g2, 256, 0, stream>>>(comb, i2o, out, partials);

    rnn_reduce<<<1, SEQ, 0, stream>>>(partials, invs);

    dim3 g4(BATCH * VOCAB / (256 * 4), SEQ);    // (2000, 128)
    rnn_scale<<<g4, 256, 0, stream>>>(out, invs);
}